// GCNLayer_23210003267965
// MI455X (gfx1250) — compile-verified
//
#include <hip/hip_runtime.h>

typedef __attribute__((ext_vector_type(2))) float v2f;
typedef __attribute__((ext_vector_type(4))) float v4f;
typedef __attribute__((ext_vector_type(8))) float v8f;

#define N_NODES 50000
#define N_EDGES 800000
#define D 128   // D_IN == D_OUT == 128

// ---------------------------------------------------------------------------
// Kernel 1: zero the SpMM accumulator f in workspace (b128 stores).
// ---------------------------------------------------------------------------
__global__ void gcn_zero_kernel(v4f* __restrict__ p, int n4) {
    int i = blockIdx.x * blockDim.x + threadIdx.x;
    if (i < n4) {
        v4f z = {};
        p[i] = z;
    }
}

// ---------------------------------------------------------------------------
// Kernel 2: SpMM scatter. One wave32 per edge. Edge metadata is forced
// wave-uniform via readfirstlane so row/col/val become scalar (SMEM) loads;
// each lane gathers 4 contiguous feature dims (global_load_b128) and issues
// 4 hardware global_atomic_add_f32 into the L2-resident accumulator f.
// ---------------------------------------------------------------------------
__global__ void __launch_bounds__(256)
gcn_spmm_kernel(const int* __restrict__ adj_row,
                const int* __restrict__ adj_col,
                const float* __restrict__ adj_vals,
                const float* __restrict__ features,
                float* __restrict__ f) {
    const int waveInBlk = __builtin_amdgcn_readfirstlane(threadIdx.x >> 5);
    const int lane      = threadIdx.x & 31;
    const int edge      = blockIdx.x * 8 + waveInBlk;   // wave-uniform (SGPR)
    if (edge >= N_EDGES) return;

    const int   r = adj_row[edge];    // scalar loads (uniform index)
    const int   c = adj_col[edge];
    const float v = adj_vals[edge];

    const v4f x = *(const v4f*)(features + (size_t)c * D + lane * 4);
    float* dst = f + (size_t)r * D + lane * 4;

    unsafeAtomicAdd(dst + 0, v * x.x);
    unsafeAtomicAdd(dst + 1, v * x.y);
    unsafeAtomicAdd(dst + 2, v * x.z);
    unsafeAtomicAdd(dst + 3, v * x.w);
}

// ---------------------------------------------------------------------------
// Kernel 3: out = f @ W^T + b via V_WMMA_F32_16X16X4_F32 (full fp32).
// One wave computes one 16x16 output tile (M = nodes, N = out dims), K = 128.
//
// K-slot mapping (our choice, identical for A and B, sum over K commutes):
//   pair p (p = 0..15) covers real k = 8p..8p+7.
//   half = lane>>4 lanes own real k = 8p + 4*half .. 8p + 4*half + 3
//   -> ONE global_load_b128 per matrix per pair; .xy feeds WMMA#1
//      (slots K0/K1 from lanes 0-15, K2/K3 from lanes 16-31), .zw feeds WMMA#2.
//   WMMA#1 accumulates k {8p,8p+1,8p+4,8p+5}; WMMA#2 {8p+2,8p+3,8p+6,8p+7}.
//
// C/D (16x16 f32): VGPR r -> out[m0 + r + 8*half][o0 + l]
// ---------------------------------------------------------------------------
__global__ void __launch_bounds__(256)
gcn_gemm_wmma_kernel(const float* __restrict__ f,
                     const float* __restrict__ W,
                     const float* __restrict__ bias,
                     float* __restrict__ out) {
    const int waveInBlk = threadIdx.x >> 5;
    const int lane      = threadIdx.x & 31;
    const int waveId    = blockIdx.x * 8 + waveInBlk;

    const int mTile = waveId >> 3;   // 0..3124  (node tile)
    const int oTile = waveId & 7;    // 0..7     (output-dim tile)
    const int half  = lane >> 4;     // 0 or 1
    const int l     = lane & 15;

    const int m0 = mTile * 16;
    const int o0 = oTile * 16;

    // 16B-aligned per-lane bases (rows are 512B aligned, offset 4*half floats)
    const float* aRow = f + (size_t)(m0 + l) * D + 4 * half;  // A: f rows
    const float* bRow = W + (size_t)(o0 + l) * D + 4 * half;  // B: W rows

    v8f acc = {};

#pragma unroll
    for (int p = 0; p < 16; ++p) {
        const v4f va = *(const v4f*)(aRow + 8 * p);
        const v4f vb = *(const v4f*)(bRow + 8 * p);
        const v2f a0 = {va.x, va.y};
        const v2f b0 = {vb.x, vb.y};
        const v2f a1 = {va.z, va.w};
        const v2f b1 = {vb.z, vb.w};
        // 8 args: (neg_a, A, neg_b, B, c_mod, C, reuse_a, reuse_b)
        acc = __builtin_amdgcn_wmma_f32_16x16x4_f32(
            false, a0, false, b0, (short)0, acc, false, false);
        acc = __builtin_amdgcn_wmma_f32_16x16x4_f32(
            false, a1, false, b1, (short)0, acc, false, false);
    }

    const float bv = bias[o0 + l];
    float* outBase = out + (size_t)(m0 + 8 * half) * D + o0 + l;
#pragma unroll
    for (int r = 0; r < 8; ++r) {
        outBase[(size_t)r * D] = acc[r] + bv;
    }
}

// ---------------------------------------------------------------------------
extern "C" void kernel_launch(void* const* d_in, const int* in_sizes, int n_in,
                              void* d_out, int out_size, void* d_ws, size_t ws_size,
                              hipStream_t stream) {
    const int*   adj_row  = (const int*)d_in[0];
    const int*   adj_col  = (const int*)d_in[1];
    const float* adj_vals = (const float*)d_in[2];
    const float* features = (const float*)d_in[3];
    const float* W        = (const float*)d_in[4];
    const float* bias     = (const float*)d_in[5];
    float*       out      = (float*)d_out;

    float* f = (float*)d_ws;                  // N_NODES * D floats = 25.6 MB scratch

    // 1) zero accumulator
    const int n4 = N_NODES * D / 4;           // 1.6M float4 stores
    gcn_zero_kernel<<<(n4 + 255) / 256, 256, 0, stream>>>((v4f*)f, n4);

    // 2) SpMM scatter: one wave32 per edge, 8 edges per 256-thread block
    gcn_spmm_kernel<<<(N_EDGES + 7) / 8, 256, 0, stream>>>(
        adj_row, adj_col, adj_vals, features, f);

    // 3) WMMA GEMM: (50000/16) m-tiles * 8 o-tiles = 25000 waves / 8 per block
    const int gemmBlocks = (N_NODES / 16) * 8 / 8;   // 3125
    gcn_gemm_wmma_kernel<<<gemmBlocks, 256, 0, stream>>>(f, W, bias, out);
}